// VectorQuantizer_10892037063058
// MI455X (gfx1250) — compile-verified
//
#include <hip/hip_runtime.h>
#include <math.h>
#include <stdint.h>

// ---------------------------------------------------------------------------
// VQ-VAE vector quantizer forward, MI455X (gfx1250), wave32 + WMMA fp32 + TDM.
//
// Sizes: z (16,64,64,64) f32 -> z_flat (N=65536, D=64); codebook (K=1024, 64).
// distances = |z|^2 + |c|^2 - 2 z.c^T  : N x K GEMM done with
// V_WMMA_F32_16X16X4_F32 (exact fp32, matches reference argmin semantics).
// Codebook (256 KB) staged into LDS via TENSOR_LOAD_TO_LDS (Tensor Data
// Mover), waited with s_wait_tensorcnt.
// ---------------------------------------------------------------------------

#define K_CODES  1024
#define DIM      64
#define HW       4096          // 64*64
#define N_VEC    65536         // 16*64*64
#define NUMEL    4194304       // N_VEC * DIM

typedef __attribute__((ext_vector_type(2))) float v2f;
typedef __attribute__((ext_vector_type(8))) float v8f;
typedef __attribute__((ext_vector_type(4))) unsigned int v4u;
typedef __attribute__((ext_vector_type(8))) int v8i;
typedef __attribute__((ext_vector_type(4))) int v4i;

// d_out layout (float elements), reference tuple order:
//   z_quantized_st[4194304], loss[1], perplexity[1],
//   encodings[67108864], encoding_indices[65536] (as float), distances[67108864]
#define OFF_ZQ    ((size_t)0)
#define OFF_LOSS  ((size_t)4194304)
#define OFF_PERP  ((size_t)4194305)
#define OFF_ENC   ((size_t)4194306)
#define OFF_IDX   ((size_t)71303170)
#define OFF_DIST  ((size_t)71368706)

// workspace layout (bytes):
//   [0)      int   ws_idx[65536]
//   [262144) uint  counts[1024]
//   [266240) float loss_sum
#define WS_COUNTS_OFF 262144
#define WS_LOSS_OFF   266240

// ---------------------------------------------------------------------------
// init: zero histogram + loss accumulator (must be re-done every call)
// ---------------------------------------------------------------------------
__global__ void vq_init(unsigned int* __restrict__ counts,
                        float* __restrict__ loss_sum) {
  int t = threadIdx.x;
  if (t < K_CODES) counts[t] = 0u;
  if (t == 0) *loss_sum = 0.0f;
}

// ---------------------------------------------------------------------------
// Main kernel: distances GEMM (WMMA fp32) + argmin + distance output.
// 512 blocks x 256 threads (8 waves). Wave w handles rows blk*128 + w*16.
// Dynamic LDS: codebook (65536 f) + per-code squared norms (1024 f) = 260 KB.
// ---------------------------------------------------------------------------
__global__ __launch_bounds__(256)
void vq_main(const float* __restrict__ z, const float* __restrict__ cb,
             float* __restrict__ dist_out, float* __restrict__ idx_out_f,
             int* __restrict__ idx_ws) {
  extern __shared__ float smem[];
  float* s_cb = smem;                 // [1024*64]  (at dynamic-LDS offset 0)
  float* s_c2 = smem + K_CODES * DIM; // [1024]

  const int tid = threadIdx.x;

#if __has_builtin(__builtin_amdgcn_tensor_load_to_lds) && \
    __has_builtin(__builtin_amdgcn_s_wait_tensorcnt)
  // ---- stage whole codebook into LDS with the Tensor Data Mover ----
  // One TDM op issued by wave 0 only (TDM ignores EXEC; per-wave issue).
  // D# per CDNA5 ISA: 2D tensor, 4B elements,
  // tensor = tile = 64 x 1024, row stride 64 elements, dest LDS offset 0.
  // This toolchain's builtin takes 6 operands:
  //   (v4u group0, v8i group1, v4i group2, v4i group3, v8i extra, i32 cpol)
  if (tid < 32) {
    unsigned long long ga = (unsigned long long)(uintptr_t)cb;
    v4u g0;
    g0.x = 1u;                                   // count=1, user descriptor
    g0.y = 0u;                                   // lds_addr = 0 (bytes)
    g0.z = (unsigned)(ga & 0xFFFFFFFFull);       // global_addr[31:0]
    g0.w = (unsigned)((ga >> 32) & 0x1FFFFFFull) // global_addr[56:32]
           | (2u << 30);                         // type = 2 ("image")
    v8i g1;
    g1[0] = (int)(2u << 16);     // wg_mask=0, data_size=2 (4 bytes)
    g1[1] = (int)(64u << 16);    // tensor_dim0[15:0]=64 (bits 63:48)
    g1[2] = (int)(1024u << 16);  // tensor_dim0 hi=0; tensor_dim1[15:0]=1024
    g1[3] = (int)(64u << 16);    // tensor_dim1 hi=0; tile_dim0=64
    g1[4] = 1024;                // tile_dim1=1024; tile_dim2=0
    g1[5] = 64;                  // tensor_dim0_stride[31:0]=64
    g1[6] = 0;                   // stride hi + tensor_dim1_stride lo (unused)
    g1[7] = 0;
    v4i gz4 = {0, 0, 0, 0};      // groups 2/3 unused for 2D tensor
    v8i gz8 = {0, 0, 0, 0, 0, 0, 0, 0};
    __builtin_amdgcn_tensor_load_to_lds(g0, g1, gz4, gz4, gz8, 0);
    __builtin_amdgcn_s_wait_tensorcnt(0);
  }
  __syncthreads();
#else
  // ---- fallback: cooperative codebook load (float4, coalesced) ----
  {
    const float4* cb4 = (const float4*)cb;
    float4* s_cb4 = (float4*)s_cb;
#pragma unroll 4
    for (int it = 0; it < 64; ++it) {
      int i4 = it * 256 + tid;     // 16384 float4 total
      s_cb4[i4] = cb4[i4];
    }
  }
  __syncthreads();
#endif

  // ---- per-code squared norms ----
  for (int k = tid; k < K_CODES; k += 256) {
    const float* r = s_cb + k * DIM;
    float s = 0.0f;
#pragma unroll
    for (int d = 0; d < DIM; ++d) s += r[d] * r[d];
    s_c2[k] = s;
  }
  __syncthreads();

  const int wave = tid >> 5;
  const int lane = tid & 31;
  const int l15  = lane & 15;
  const int hi   = lane >> 4;         // 0: lanes 0-15, 1: lanes 16-31
  const int row_base = blockIdx.x * 128 + wave * 16;

  // ---- load A fragments (ISA 32-bit A 16x4 layout):
  // VGPR0 = K {0 | 2}, VGPR1 = K {1 | 3} split at lane 16.
  // z_flat[n,d] = z[b*262144 + d*4096 + hw], n = b*4096 + hw.
  const int n_lane = row_base + l15;
  const int bb = n_lane >> 12;
  const int hw = n_lane & 4095;
  const float* zp = z + ((size_t)bb << 18) + (size_t)hw;

  v2f a[16];
  float z2p = 0.0f;
#pragma unroll
  for (int t = 0; t < 16; ++t) {
    const int d0 = 4 * t + 2 * hi;
    float x = zp[(size_t)d0 << 12];
    float y = zp[(size_t)(d0 + 1) << 12];
    a[t].x = x;
    a[t].y = y;
    z2p += x * x + y * y;             // half the dims live in this half-wave
  }
  // combine halves: lane L ends up with |z_row(L%16)|^2
  float z2row = z2p + __shfl_xor(z2p, 16, 32);
  // broadcast the 8 row-norms this half-wave's C fragment needs
  float zz[8];
#pragma unroll
  for (int i = 0; i < 8; ++i) zz[i] = __shfl(z2row, i + 8 * hi, 32);

  float minv[8];
  int   mink[8];
#pragma unroll
  for (int i = 0; i < 8; ++i) { minv[i] = 3.4e38f; mink[i] = 0; }

  // ---- K loop: 64 column tiles of 16 codes ----
  for (int ct = 0; ct < 64; ++ct) {
    const int col = ct * 16 + l15;
    // B fragment source: contiguous (d, d+1) pair of code 'col' in LDS.
    const float* bcol = s_cb + col * DIM + 2 * hi;
    v8f acc = {};
#pragma unroll
    for (int t = 0; t < 16; ++t) {
      v2f bf;
      bf.x = bcol[4 * t];             // B rows {0|2} of this 4-dim slab
      bf.y = bcol[4 * t + 1];         // B rows {1|3}
      acc = __builtin_amdgcn_wmma_f32_16x16x4_f32(
          false, a[t], false, bf, (short)0, acc, false, false);
    }
    const float c2v = s_c2[col];
    // C layout: VGPR i = rows {i | i+8}, N = lane%16
    float* drow = dist_out + (size_t)(row_base + 8 * hi) * K_CODES + col;
#pragma unroll
    for (int i = 0; i < 8; ++i) {
      float dist = zz[i] + c2v - 2.0f * acc[i];
      drow[(size_t)i * K_CODES] = dist;
      if (dist < minv[i]) { minv[i] = dist; mink[i] = col; }  // strict < keeps
    }                                                         // earliest k
  }

  // ---- argmin reduce across the 16 lanes holding one row's columns ----
#pragma unroll
  for (int i = 0; i < 8; ++i) {
    float v = minv[i];
    int   k = mink[i];
#pragma unroll
    for (int m = 8; m >= 1; m >>= 1) {
      float ov = __shfl_xor(v, m, 16);
      int   ok = __shfl_xor(k, m, 16);
      if (ov < v || (ov == v && ok < k)) { v = ov; k = ok; }  // first-index tie
    }
    mink[i] = k;
  }
  if (l15 == 0) {
#pragma unroll
    for (int i = 0; i < 8; ++i) {
      int n = row_base + i + 8 * hi;
      idx_ws[n] = mink[i];
      idx_out_f[n] = (float)mink[i];
    }
  }
}

// ---------------------------------------------------------------------------
// histogram of selected codes (for perplexity)
// ---------------------------------------------------------------------------
__global__ void vq_hist(const int* __restrict__ idx_ws,
                        unsigned int* __restrict__ counts) {
  int n = blockIdx.x * 256 + threadIdx.x;
  atomicAdd(&counts[idx_ws[n]], 1u);
}

// ---------------------------------------------------------------------------
// one-hot encodings (N x K), float4 stores. 65536 blocks x 256 threads.
// ---------------------------------------------------------------------------
__global__ void vq_onehot(const int* __restrict__ idx_ws,
                          float4* __restrict__ enc4) {
  size_t g = (size_t)blockIdx.x * 256 + threadIdx.x;  // 16,777,216 float4
  int n  = (int)(g >> 8);                             // 256 float4 per row
  int c0 = ((int)g & 255) << 2;
  int k  = idx_ws[n];
  float4 v;
  v.x = (c0 + 0 == k) ? 1.0f : 0.0f;
  v.y = (c0 + 1 == k) ? 1.0f : 0.0f;
  v.z = (c0 + 2 == k) ? 1.0f : 0.0f;
  v.w = (c0 + 3 == k) ? 1.0f : 0.0f;
  enc4[g] = v;
}

// ---------------------------------------------------------------------------
// gather codebook rows back to (B,D,H,W) layout (straight-through forward
// value == z_q) + fused MSE accumulation. loss = (beta+1)*MSE = 1.25*MSE.
// ---------------------------------------------------------------------------
__global__ __launch_bounds__(256)
void vq_quant(const float* __restrict__ z, const float* __restrict__ cb,
              const int* __restrict__ idx_ws, float* __restrict__ zq_out,
              float* __restrict__ loss_sum) {
  __shared__ float red[256];
  unsigned int g = blockIdx.x * 256 + threadIdx.x;    // < 4,194,304
  int d  = (g >> 12) & 63;
  int bb = g >> 18;
  int hw = g & 4095;
  int n  = (bb << 12) | hw;
  int k  = idx_ws[n];
  float val  = cb[k * DIM + d];
  zq_out[g] = val;
  float diff = val - z[g];
  red[threadIdx.x] = diff * diff;
  __syncthreads();
  for (int s = 128; s > 0; s >>= 1) {
    if (threadIdx.x < s) red[threadIdx.x] += red[threadIdx.x + s];
    __syncthreads();
  }
  if (threadIdx.x == 0) atomicAdd(loss_sum, red[0]);
}

// ---------------------------------------------------------------------------
// finalize: perplexity + loss scalars. 1 block x 1024 threads.
// ---------------------------------------------------------------------------
__global__ void vq_final(const unsigned int* __restrict__ counts,
                         const float* __restrict__ loss_sum,
                         float* __restrict__ out_loss,
                         float* __restrict__ out_perp) {
  __shared__ float red[1024];
  int k = threadIdx.x;
  float p = (float)counts[k] * (1.0f / (float)N_VEC);
  red[k] = p * logf(p + 1e-10f);
  __syncthreads();
  for (int s = 512; s > 0; s >>= 1) {
    if (k < s) red[k] += red[k + s];
    __syncthreads();
  }
  if (k == 0) {
    *out_perp = expf(-red[0]);
    *out_loss = 1.25f * (*loss_sum) * (1.0f / (float)NUMEL);
  }
}

// ---------------------------------------------------------------------------
extern "C" void kernel_launch(void* const* d_in, const int* in_sizes, int n_in,
                              void* d_out, int out_size, void* d_ws,
                              size_t ws_size, hipStream_t stream) {
  const float* z  = (const float*)d_in[0];
  const float* cb = (const float*)d_in[1];
  float* out = (float*)d_out;

  int*          ws_idx   = (int*)d_ws;
  unsigned int* counts   = (unsigned int*)((char*)d_ws + WS_COUNTS_OFF);
  float*        loss_sum = (float*)((char*)d_ws + WS_LOSS_OFF);

  vq_init<<<1, 1024, 0, stream>>>(counts, loss_sum);

  const size_t shmem = (size_t)(K_CODES * DIM + K_CODES) * sizeof(float); // 260KB
  static_cast<void>(hipFuncSetAttribute(
      reinterpret_cast<const void*>(vq_main),
      hipFuncAttributeMaxDynamicSharedMemorySize, (int)shmem));
  vq_main<<<N_VEC / 128, 256, shmem, stream>>>(z, cb, out + OFF_DIST,
                                               out + OFF_IDX, ws_idx);

  vq_hist<<<N_VEC / 256, 256, 0, stream>>>(ws_idx, counts);
  vq_onehot<<<65536, 256, 0, stream>>>(ws_idx, (float4*)(out + OFF_ENC));
  vq_quant<<<NUMEL / 256, 256, 0, stream>>>(z, cb, ws_idx, out + OFF_ZQ,
                                            loss_sum);
  vq_final<<<1, 1024, 0, stream>>>(counts, loss_sum, out + OFF_LOSS,
                                   out + OFF_PERP);
}